// PointPillars_trt_17678085390525
// MI455X (gfx1250) — compile-verified
//
#include <hip/hip_runtime.h>
#include <hip/hip_bf16.h>
#include <math.h>

#define N_ANCH   321408
#define PRE_MAX  4096
#define POST_MAX 300
#define CAND_CAP 8192

typedef __attribute__((ext_vector_type(2))) float v2f;
typedef __attribute__((ext_vector_type(8))) float v8f;

__device__ __forceinline__ unsigned int fkey(float f) {
    unsigned int b = __float_as_uint(f);
    return (b & 0x80000000u) ? ~b : (b | 0x80000000u);
}
__device__ __forceinline__ float funkey(unsigned int k) {
    unsigned int b = (k & 0x80000000u) ? (k ^ 0x80000000u) : ~k;
    return __uint_as_float(b);
}

// ---------------- init: zero histograms / counters ----------------
__global__ void init_kernel(unsigned int* p, int n) {
    int i = blockIdx.x * blockDim.x + threadIdx.x;
    if (i < n) p[i] = 0u;
}

// ---------------- stage 1: decode boxes + score/label/dir ----------------
__global__ void decode_score_kernel(const float* __restrict__ bp, const float* __restrict__ cp,
                                    const float* __restrict__ dp, const float* __restrict__ an,
                                    float* __restrict__ boxes, unsigned int* __restrict__ keys,
                                    int* __restrict__ lbl, int* __restrict__ dir) {
    int i = blockIdx.x * blockDim.x + threadIdx.x;
    if (i >= N_ANCH) return;
    const float* a = an + (long)i * 7;
    const float* e = bp + (long)i * 7;
    float xa = a[0], ya = a[1], za = a[2], wa = a[3], la = a[4], ha = a[5], ra = a[6];
    float xt = e[0], yt = e[1], zt = e[2], wt = e[3], lt = e[4], ht = e[5], rt = e[6];
    za += ha * 0.5f;
    float diag = sqrtf(la * la + wa * wa);
    float xg = xt * diag + xa;
    float yg = yt * diag + ya;
    float zg = zt * ha + za;
    float lg = expf(lt) * la;
    float wg = expf(wt) * wa;
    float hg = expf(ht) * ha;
    float rg = rt + ra;
    zg -= hg * 0.5f;
    float* o = boxes + (long)i * 7;
    o[0] = xg; o[1] = yg; o[2] = zg; o[3] = wg; o[4] = lg; o[5] = hg; o[6] = rg;

    float c0 = cp[i * 3 + 0], c1 = cp[i * 3 + 1], c2 = cp[i * 3 + 2];
    float m = fmaxf(c0, fmaxf(c1, c2));
    int label = (c0 >= m) ? 0 : ((c1 >= m) ? 1 : 2);          // first-max, like argmax
    float ts = 1.0f / (1.0f + expf(-m));                      // sigmoid monotone -> max/argmax carry over
    float sc = (ts >= 0.05f) ? ts : -1.0f;
    keys[i] = fkey(sc);
    lbl[i] = label;
    dir[i] = (dp[i * 2 + 1] > dp[i * 2 + 0]) ? 1 : 0;
}

// ---------------- stage 2: exact top-4096 via 2-level radix select ----------------
__global__ void hist1_kernel(const unsigned int* __restrict__ keys, unsigned int* __restrict__ hist) {
    int i = blockIdx.x * blockDim.x + threadIdx.x;
    if (i >= N_ANCH) return;
    atomicAdd(&hist[keys[i] >> 21], 1u);
}
__global__ void scan1_kernel(const unsigned int* __restrict__ hist, unsigned int* __restrict__ meta) {
    unsigned int cum = 0;
    for (int b = 2047; b >= 0; --b) {
        unsigned int c = hist[b];
        if (cum + c >= PRE_MAX) { meta[0] = (unsigned)b; meta[1] = cum; return; }
        cum += c;
    }
    meta[0] = 0; meta[1] = cum;
}
__global__ void hist2_kernel(const unsigned int* __restrict__ keys, const unsigned int* __restrict__ meta,
                             unsigned int* __restrict__ hist) {
    int i = blockIdx.x * blockDim.x + threadIdx.x;
    if (i >= N_ANCH) return;
    unsigned int k = keys[i];
    if ((k >> 21) == meta[0]) atomicAdd(&hist[(k >> 10) & 0x7FFu], 1u);
}
__global__ void scan2_kernel(const unsigned int* __restrict__ hist, unsigned int* __restrict__ meta) {
    unsigned int cum = meta[1];
    unsigned int b = meta[0];
    for (int s = 2047; s >= 0; --s) {
        unsigned int c = hist[s];
        if (cum + c >= PRE_MAX) { meta[2] = (b << 21) | ((unsigned)s << 10); return; }
        cum += c;
    }
    meta[2] = b << 21;
}
__global__ void compact_kernel(const unsigned int* __restrict__ keys, const unsigned int* __restrict__ meta,
                               unsigned long long* __restrict__ cand, unsigned int* __restrict__ cnt) {
    int i = blockIdx.x * blockDim.x + threadIdx.x;
    if (i >= N_ANCH) return;
    unsigned int k = keys[i];
    if (k >= meta[2]) {
        unsigned int p = atomicAdd(cnt, 1u);
        if (p < CAND_CAP)
            cand[p] = ((unsigned long long)k << 32) | (unsigned long long)(0xFFFFFFFFu - (unsigned)i);
    }
}
__global__ __launch_bounds__(1024)
void sort_pre_kernel(const unsigned long long* __restrict__ cand, const unsigned int* __restrict__ cnt,
                     int* __restrict__ pre_idx, float* __restrict__ pre_sc) {
    __shared__ unsigned long long s[CAND_CAP];   // 64 KB
    int t = threadIdx.x;
    unsigned int n = *cnt; if (n > CAND_CAP) n = CAND_CAP;
    for (int i = t; i < CAND_CAP; i += 1024) s[i] = (i < (int)n) ? cand[i] : 0xFFFFFFFFull;
    __syncthreads();
    for (int k = 2; k <= CAND_CAP; k <<= 1)
        for (int j = k >> 1; j > 0; j >>= 1) {
            for (int i = t; i < CAND_CAP; i += 1024) {
                int ixj = i ^ j;
                if (ixj > i) {
                    unsigned long long a = s[i], b = s[ixj];
                    bool desc = ((i & k) == 0);
                    if (desc ? (a < b) : (a > b)) { s[i] = b; s[ixj] = a; }
                }
            }
            __syncthreads();
        }
    for (int i = t; i < PRE_MAX; i += 1024) {
        unsigned long long e = s[i];
        unsigned int key = (unsigned)(e >> 32);
        unsigned int idx = 0xFFFFFFFFu - (unsigned)(e & 0xFFFFFFFFull);
        pre_idx[i] = (int)idx;
        pre_sc[i] = (i < (int)n) ? funkey(key) : -1.0f;
    }
}

// ---------------- stage 3: standup (axis-aligned) boxes + areas + valid bits ----------------
__global__ void standup_kernel(const int* __restrict__ pre_idx, const float* __restrict__ pre_sc,
                               const float* __restrict__ boxes, float4* __restrict__ nbox,
                               float* __restrict__ area, unsigned int* __restrict__ validW) {
    int t = blockIdx.x * 256 + threadIdx.x;   // 0..4095
    int idx = pre_idx[t];
    const float* b = boxes + (long)idx * 7;
    float x = b[0], y = b[1], w = b[3], l = b[4], r = b[6];
    float ca = fabsf(__cosf(r)), sa = fabsf(__sinf(r));
    float ex = 0.5f * (ca * w + sa * l);
    float ey = 0.5f * (sa * w + ca * l);
    nbox[t] = make_float4(x - ex, y - ey, x + ex, y + ey);
    area[t] = 4.0f * ex * ey;
    unsigned int bal = __builtin_amdgcn_ballot_w32(pre_sc[t] > 0.0f);
    if ((threadIdx.x & 31) == 0) validW[t >> 5] = bal;
}

// ---------------- stage 4: 4096x4096 IoU -> suppression bitmask (WMMA tile) ----------------
// Each wave owns a 16x32 tile. V_WMMA_F32_16X16X4_F32 computes D[i][j] = area_i + area_j + eps
// (A=[area_i,1,0,0], B=[1;area_j;0;0], C=eps). Then iou>0.5  <=>  3*inter > D  (no division).
__global__ __launch_bounds__(256)
void iou_sup_kernel(const float4* __restrict__ nbox, const float* __restrict__ area,
                    unsigned int* __restrict__ sup) {
    int wid  = blockIdx.x * 8 + (threadIdx.x >> 5);   // 0..32767
    int lane = threadIdx.x & 31;
    int ct = wid & 127;        // col tile of 32
    int rt = wid >> 7;         // row tile of 16
    int i0 = rt << 4;
    int j0 = ct << 5;

    int n = lane & 15;
    bool lo = lane < 16;

    v2f a, b0, b1;
    a.x  = lo ? area[i0 + n] : 0.0f;   a.y  = lo ? 1.0f : 0.0f;          // A: 16x4, rows=area_i,1,0,0
    b0.x = lo ? 1.0f : 0.0f;           b0.y = lo ? area[j0 + n] : 0.0f;  // B: 4x16, cols=1,area_j,0,0
    b1.x = b0.x;                       b1.y = lo ? area[j0 + 16 + n] : 0.0f;
    v8f c = {1e-8f, 1e-8f, 1e-8f, 1e-8f, 1e-8f, 1e-8f, 1e-8f, 1e-8f};

    v8f d0 = __builtin_amdgcn_wmma_f32_16x16x4_f32(false, a, false, b0, (short)0, c, false, false);
    v8f d1 = __builtin_amdgcn_wmma_f32_16x16x4_f32(false, a, false, b1, (short)0, c, false, false);

    float4 cb0 = nbox[j0 + n];
    float4 cb1 = nbox[j0 + 16 + n];
    int rbase = i0 + (lo ? 0 : 8);     // accumulator VGPR r holds row r (lanes 0-15) / r+8 (lanes 16-31)

    unsigned int wlow[8], whigh[8];
#pragma unroll
    for (int r = 0; r < 8; ++r) {
        float4 rb = nbox[rbase + r];
        float ix0 = fmaxf(0.0f, fminf(rb.z, cb0.z) - fmaxf(rb.x, cb0.x));
        float iy0 = fmaxf(0.0f, fminf(rb.w, cb0.w) - fmaxf(rb.y, cb0.y));
        float ix1 = fmaxf(0.0f, fminf(rb.z, cb1.z) - fmaxf(rb.x, cb1.x));
        float iy1 = fmaxf(0.0f, fminf(rb.w, cb1.w) - fmaxf(rb.y, cb1.y));
        bool p0 = 3.0f * (ix0 * iy0) > d0[r];
        bool p1 = 3.0f * (ix1 * iy1) > d1[r];
        unsigned int bal0 = __builtin_amdgcn_ballot_w32(p0);
        unsigned int bal1 = __builtin_amdgcn_ballot_w32(p1);
        wlow[r]  = (bal0 & 0xFFFFu) | (bal1 << 16);       // row r      : cols j0..j0+31
        whigh[r] = (bal0 >> 16) | (bal1 & 0xFFFF0000u);   // row r+8
    }
    if (lane == 0) {
        unsigned int* rowp = sup + ct;
#pragma unroll
        for (int r = 0; r < 8; ++r) {
            rowp[(i0 + r) * 128]     = wlow[r];
            rowp[(i0 + 8 + r) * 128] = whigh[r];
        }
    }
}

// ---------------- stage 5: greedy NMS, single wave32, bitset scan ----------------
__global__ void nms_kernel(const unsigned int* __restrict__ sup, const unsigned int* __restrict__ validW,
                           unsigned int* __restrict__ keepW) {
    int lane = threadIdx.x;   // 32 lanes; lane owns words lane, lane+32, lane+64, lane+96
    unsigned int s0 = 0, s1 = 0, s2 = 0, s3 = 0;
    unsigned int k0 = 0, k1 = 0, k2 = 0, k3 = 0;
    unsigned int v0 = validW[lane], v1 = validW[lane + 32], v2 = validW[lane + 64], v3 = validW[lane + 96];
    for (int i = 0; i < PRE_MAX; ++i) {
        int w = i >> 5;
        int owner = w & 31;
        int reg = w >> 5;
        unsigned int sw = (reg == 0) ? s0 : (reg == 1) ? s1 : (reg == 2) ? s2 : s3;
        unsigned int vw = (reg == 0) ? v0 : (reg == 1) ? v1 : (reg == 2) ? v2 : v3;
        unsigned int swb = __shfl(sw, owner, 32);
        unsigned int vwb = __shfl(vw, owner, 32);
        unsigned int bit = 1u << (i & 31);
        bool keep = (vwb & bit) && !(swb & bit);     // uniform across wave
        if (keep) {
            const unsigned int* row = sup + (long)i * 128;
            s0 |= row[lane]; s1 |= row[lane + 32]; s2 |= row[lane + 64]; s3 |= row[lane + 96];
            if (lane == owner) {
                if (reg == 0) k0 |= bit; else if (reg == 1) k1 |= bit;
                else if (reg == 2) k2 |= bit; else k3 |= bit;
            }
        }
    }
    keepW[lane] = k0; keepW[lane + 32] = k1; keepW[lane + 64] = k2; keepW[lane + 96] = k3;
}

// ---------------- stage 6: top-300 of kept + finalize outputs ----------------
__global__ __launch_bounds__(1024)
void topk_final_kernel(const float* __restrict__ pre_sc, const int* __restrict__ pre_idx,
                       const unsigned int* __restrict__ keepW, const float* __restrict__ boxes,
                       const int* __restrict__ lbl, const int* __restrict__ dir,
                       float* __restrict__ out) {
    __shared__ unsigned long long s[PRE_MAX];   // 32 KB
    int t = threadIdx.x;
    for (int i = t; i < PRE_MAX; i += 1024) {
        bool kept = (keepW[i >> 5] >> (i & 31)) & 1u;
        float v = kept ? pre_sc[i] : -1.0f;
        s[i] = ((unsigned long long)fkey(v) << 32) | (unsigned long long)(0xFFFFFFFFu - (unsigned)i);
    }
    __syncthreads();
    for (int k = 2; k <= PRE_MAX; k <<= 1)
        for (int j = k >> 1; j > 0; j >>= 1) {
            for (int i = t; i < PRE_MAX; i += 1024) {
                int ixj = i ^ j;
                if (ixj > i) {
                    unsigned long long a = s[i], b = s[ixj];
                    bool desc = ((i & k) == 0);
                    if (desc ? (a < b) : (a > b)) { s[i] = b; s[ixj] = a; }
                }
            }
            __syncthreads();
        }
    if (t < POST_MAX) {
        unsigned long long e = s[t];
        float psc = funkey((unsigned)(e >> 32));
        int pi = (int)(0xFFFFFFFFu - (unsigned)(e & 0xFFFFFFFFull));
        bool v = psc > 0.0f;
        int sel = pre_idx[pi];
        const float* bb = boxes + (long)sel * 7;
        float bx[7];
#pragma unroll
        for (int q = 0; q < 7; ++q) bx[q] = bb[q];
        int dl = dir[sel];
        bool opp = (bx[6] > 0.0f) != (dl > 0);
        bx[6] += opp ? 3.14159265358979323846f : 0.0f;
#pragma unroll
        for (int q = 0; q < 7; ++q) out[t * 7 + q] = v ? bx[q] : 0.0f;
        out[2100 + t] = v ? psc : 0.0f;
        out[2400 + t] = v ? (float)lbl[sel] : -1.0f;
        out[2700 + t] = v ? 1.0f : 0.0f;
    }
}

extern "C" void kernel_launch(void* const* d_in, const int* in_sizes, int n_in,
                              void* d_out, int out_size, void* d_ws, size_t ws_size,
                              hipStream_t stream) {
    const float* box_preds = (const float*)d_in[0];
    const float* cls_preds = (const float*)d_in[1];
    const float* dir_preds = (const float*)d_in[2];
    const float* anchors   = (const float*)d_in[3];
    float* out = (float*)d_out;

    char* ws = (char*)d_ws;
    size_t off = 0;
    auto alloc = [&](size_t bytes) -> void* {
        void* p = ws + off;
        off += bytes;
        off = (off + 255) & ~(size_t)255;
        return p;
    };
    float*              boxes  = (float*)alloc((size_t)N_ANCH * 7 * 4);
    unsigned int*       keys   = (unsigned int*)alloc((size_t)N_ANCH * 4);
    int*                lbl    = (int*)alloc((size_t)N_ANCH * 4);
    int*                dir    = (int*)alloc((size_t)N_ANCH * 4);
    unsigned int*       zr     = (unsigned int*)alloc(4128 * 4);   // hist1|hist2|meta|cnt
    unsigned int*       hist1  = zr;
    unsigned int*       hist2  = zr + 2048;
    unsigned int*       meta   = zr + 4096;
    unsigned int*       cnt    = zr + 4112;
    unsigned long long* cand   = (unsigned long long*)alloc((size_t)CAND_CAP * 8);
    int*                preIdx = (int*)alloc((size_t)PRE_MAX * 4);
    float*              preSc  = (float*)alloc((size_t)PRE_MAX * 4);
    float4*             nbox   = (float4*)alloc((size_t)PRE_MAX * 16);
    float*              area   = (float*)alloc((size_t)PRE_MAX * 4);
    unsigned int*       validW = (unsigned int*)alloc(128 * 4);
    unsigned int*       keepW  = (unsigned int*)alloc(128 * 4);
    unsigned int*       sup    = (unsigned int*)alloc((size_t)PRE_MAX * 128 * 4);  // 2 MB bitmask

    const int NB = (N_ANCH + 255) / 256;

    init_kernel<<<(4128 + 255) / 256, 256, 0, stream>>>(zr, 4128);
    decode_score_kernel<<<NB, 256, 0, stream>>>(box_preds, cls_preds, dir_preds, anchors,
                                                boxes, keys, lbl, dir);
    hist1_kernel<<<NB, 256, 0, stream>>>(keys, hist1);
    scan1_kernel<<<1, 1, 0, stream>>>(hist1, meta);
    hist2_kernel<<<NB, 256, 0, stream>>>(keys, meta, hist2);
    scan2_kernel<<<1, 1, 0, stream>>>(hist2, meta);
    compact_kernel<<<NB, 256, 0, stream>>>(keys, meta, cand, cnt);
    sort_pre_kernel<<<1, 1024, 0, stream>>>(cand, cnt, preIdx, preSc);
    standup_kernel<<<PRE_MAX / 256, 256, 0, stream>>>(preIdx, preSc, boxes, nbox, area, validW);
    iou_sup_kernel<<<4096, 256, 0, stream>>>(nbox, area, sup);   // 32768 waves, 16x32 tiles
    nms_kernel<<<1, 32, 0, stream>>>(sup, validW, keepW);
    topk_final_kernel<<<1, 1024, 0, stream>>>(preSc, preIdx, keepW, boxes, lbl, dir, out);
}